// CapsuleBase_49039936586329
// MI455X (gfx1250) — compile-verified
//
#include <hip/hip_runtime.h>
#include <hip/hip_bf16.h>
#include <math.h>

typedef __attribute__((ext_vector_type(2))) float v2f;
typedef __attribute__((ext_vector_type(8))) float v8f;

#define N_ENT     200000
#define INIT_DIM  128
#define K_FACTORS 4
#define GCN_DIM   128
#define KD        512      // K_FACTORS * GCN_DIM
#define BATCH     4096

// ---------------------------------------------------------------------------
// Kernel 1: x = tanh(init_embed @ W_pca + b_pca)   [200000,128]@[128,512]
// One block = one 16-row M tile. 8 waves; wave w covers columns [w*64, w*64+64).
// A tile staged in LDS (padded stride); B loaded coalesced from L2-hot W.
// f32 WMMA 16x16x4: A frag = float2 (lanes 0-15: K=k,k+1 ; lanes 16-31: K=k+2,k+3),
// B frag = float2 per N tile, C = v8f per the 16x16 f32 layout.
// ---------------------------------------------------------------------------
__global__ __launch_bounds__(256)
void pca_tanh_gemm(const float* __restrict__ A,    // [N_ENT, 128]
                   const float* __restrict__ W,    // [128, 512]
                   const float* __restrict__ bias, // [512]
                   float* __restrict__ X)          // [N_ENT, 512]
{
    __shared__ float sA[16][INIT_DIM + 4];   // pad to break bank conflicts
    const int m0  = blockIdx.x * 16;
    const int tid = threadIdx.x;

    // cooperative load of the 16x128 A tile (each thread: 8 floats)
    {
        const int r = tid >> 4;            // 0..15
        const int c = (tid & 15) * 8;      // 0..120
        const float4* src = (const float4*)(A + (size_t)(m0 + r) * INIT_DIM + c);
        float4 v0 = src[0];
        float4 v1 = src[1];
        float* dst = &sA[r][c];
        dst[0] = v0.x; dst[1] = v0.y; dst[2] = v0.z; dst[3] = v0.w;
        dst[4] = v1.x; dst[5] = v1.y; dst[6] = v1.z; dst[7] = v1.w;
    }
    __syncthreads();

    const int wave  = tid >> 5;
    const int lane  = tid & 31;
    const int lh    = lane >> 4;    // half-wave select
    const int ll    = lane & 15;
    const int nbase = wave * 64;

    v8f acc[4] = {};

    for (int k = 0; k < INIT_DIM; k += 4) {
        const int ka = k + lh * 2;
        v2f a;
        a.x = sA[ll][ka + 0];
        a.y = sA[ll][ka + 1];

        const float* wp0 = W + (size_t)(ka + 0) * KD + nbase + ll;
        const float* wp1 = W + (size_t)(ka + 1) * KD + nbase + ll;
        v2f b0, b1, b2, b3;
        b0.x = wp0[ 0]; b0.y = wp1[ 0];
        b1.x = wp0[16]; b1.y = wp1[16];
        b2.x = wp0[32]; b2.y = wp1[32];
        b3.x = wp0[48]; b3.y = wp1[48];

        acc[0] = __builtin_amdgcn_wmma_f32_16x16x4_f32(false, a, false, b0, (short)0, acc[0], false, false);
        acc[1] = __builtin_amdgcn_wmma_f32_16x16x4_f32(false, a, false, b1, (short)0, acc[1], false, false);
        acc[2] = __builtin_amdgcn_wmma_f32_16x16x4_f32(false, a, false, b2, (short)0, acc[2], false, false);
        acc[3] = __builtin_amdgcn_wmma_f32_16x16x4_f32(false, a, false, b3, (short)0, acc[3], false, false);
    }

    // epilogue: bias + tanh, store per C layout (VGPR r -> rows r / r+8)
    #pragma unroll
    for (int t = 0; t < 4; ++t) {
        const int n = nbase + t * 16 + ll;
        const float bv = bias[n];
        #pragma unroll
        for (int r = 0; r < 8; ++r) {
            const int m = m0 + r + lh * 8;
            X[(size_t)m * KD + n] = tanhf(acc[t][r] + bv);
        }
    }
}

// ---------------------------------------------------------------------------
// Kernel 2: sub_emb[i,:] = x[sub[i],:] ; rel_emb[i,:] = tile(init_rel[rel[i],:],4)
// ---------------------------------------------------------------------------
__global__ __launch_bounds__(128)
void gather_kernel(const float* __restrict__ X,        // [N_ENT, 512]
                   const float* __restrict__ R,        // [N_REL2, 128]
                   const int* __restrict__ sub,
                   const int* __restrict__ rel,
                   float* __restrict__ sub_emb,        // [BATCH, 512]
                   float* __restrict__ rel_emb)        // [BATCH, 512]
{
    const int i = blockIdx.x;
    const int t = threadIdx.x;          // 0..127, one float4 each
    const int s = sub[i];
    const int r = rel[i];

    const float4* xs = (const float4*)(X + (size_t)s * KD);
    float4* so = (float4*)(sub_emb + (size_t)i * KD);
    so[t] = xs[t];

    const float4* rs = (const float4*)(R + (size_t)r * GCN_DIM);
    float4* ro = (float4*)(rel_emb + (size_t)i * KD);
    ro[t] = rs[t & 31];                 // tile 128-float row 4x
}

// ---------------------------------------------------------------------------
// Kernel 3: kfactor[k] = Z_k^T @ Z_k, Z_k = sub_emb[:, k*128:(k+1)*128]
// One wave per 16x16 output tile; K loop over the 4096 batch in steps of 4.
// ---------------------------------------------------------------------------
__global__ __launch_bounds__(32)
void gram_kernel(const float* __restrict__ S,   // sub_emb [BATCH, 512]
                 float* __restrict__ KF)        // [4, 128, 128]
{
    const int k  = blockIdx.z;
    const int d0 = blockIdx.x * 16;     // M (rows of kfactor)
    const int f0 = blockIdx.y * 16;     // N (cols of kfactor)
    const int lane = threadIdx.x;
    const int lh = lane >> 4;
    const int ll = lane & 15;

    const float* base = S + k * GCN_DIM;
    v8f acc = {};

    for (int b = 0; b < BATCH; b += 4) {
        const int br = b + lh * 2;
        const float* r0 = base + (size_t)(br + 0) * KD;
        const float* r1 = base + (size_t)(br + 1) * KD;
        v2f a, bb;
        a.x  = r0[d0 + ll];  a.y  = r1[d0 + ll];   // A = Z^T : row d, cols b
        bb.x = r0[f0 + ll];  bb.y = r1[f0 + ll];   // B = Z   : row b, cols f
        acc = __builtin_amdgcn_wmma_f32_16x16x4_f32(false, a, false, bb, (short)0, acc, false, false);
    }

    #pragma unroll
    for (int r = 0; r < 8; ++r) {
        const int d = d0 + r + lh * 8;
        KF[(size_t)k * GCN_DIM * GCN_DIM + (size_t)d * GCN_DIM + f0 + ll] = acc[r];
    }
}

// ---------------------------------------------------------------------------
// Kernel 4: M_k = (I - ones/n) @ kfactor[k]  => M_k[d,f] = K_k[d,f] - colmean_k[f]
//           loss = sum_{i<j} sum_{d,f} M_i[d,f] * M_j[f,d]
// ---------------------------------------------------------------------------
__global__ __launch_bounds__(256)
void hsic_loss_kernel(const float* __restrict__ KF, float* __restrict__ out_loss)
{
    __shared__ float cm[K_FACTORS][GCN_DIM];
    __shared__ float red[256];
    const int tid = threadIdx.x;

    for (int idx = tid; idx < K_FACTORS * GCN_DIM; idx += 256) {
        const int k = idx >> 7, f = idx & 127;
        float s = 0.f;
        const float* col = KF + (size_t)k * GCN_DIM * GCN_DIM + f;
        for (int d = 0; d < GCN_DIM; ++d) s += col[(size_t)d * GCN_DIM];
        cm[k][f] = s * (1.0f / GCN_DIM);
    }
    __syncthreads();

    float partial = 0.f;
    for (int pos = tid; pos < GCN_DIM * GCN_DIM; pos += 256) {
        const int d = pos >> 7, f = pos & 127;
        float mi[K_FACTORS], mjT[K_FACTORS];
        #pragma unroll
        for (int kk = 0; kk < K_FACTORS; ++kk) {
            mi[kk]  = KF[(size_t)kk * GCN_DIM * GCN_DIM + (size_t)d * GCN_DIM + f] - cm[kk][f];
            mjT[kk] = KF[(size_t)kk * GCN_DIM * GCN_DIM + (size_t)f * GCN_DIM + d] - cm[kk][d];
        }
        #pragma unroll
        for (int i = 0; i < K_FACTORS; ++i)
            #pragma unroll
            for (int j = i + 1; j < K_FACTORS; ++j)
                partial += mi[i] * mjT[j];
    }

    red[tid] = partial;
    __syncthreads();
    for (int s = 128; s > 0; s >>= 1) {
        if (tid < s) red[tid] += red[tid + s];
        __syncthreads();
    }
    if (tid == 0) out_loss[0] = red[0];
}

// ---------------------------------------------------------------------------
extern "C" void kernel_launch(void* const* d_in, const int* in_sizes, int n_in,
                              void* d_out, int out_size, void* d_ws, size_t ws_size,
                              hipStream_t stream)
{
    const float* init_embed = (const float*)d_in[0];
    const float* W_pca      = (const float*)d_in[1];
    const float* b_pca      = (const float*)d_in[2];
    const float* init_rel   = (const float*)d_in[3];
    const int*   sub        = (const int*)d_in[4];
    const int*   rel        = (const int*)d_in[5];

    float* out     = (float*)d_out;
    float* sub_emb = out;                                // [4096, 512]
    float* rel_emb = out + (size_t)BATCH * KD;           // [4096, 512]
    float* X       = out + (size_t)2 * BATCH * KD;       // [200000, 512]
    float* loss    = out + ((size_t)out_size - 1);       // scalar
    float* KF      = (float*)d_ws;                       // [4,128,128] = 256KB

    pca_tanh_gemm<<<dim3(N_ENT / 16), dim3(256), 0, stream>>>(init_embed, W_pca, b_pca, X);
    gather_kernel<<<dim3(BATCH), dim3(128), 0, stream>>>(X, init_rel, sub, rel, sub_emb, rel_emb);
    gram_kernel<<<dim3(8, 8, K_FACTORS), dim3(32), 0, stream>>>(sub_emb, KF);
    hsic_loss_kernel<<<dim3(1), dim3(256), 0, stream>>>(KF, loss);
}